// TrimmedConvolution_72894184948182
// MI455X (gfx1250) — compile-verified
//
#include <hip/hip_runtime.h>

typedef float v2f __attribute__((ext_vector_type(2)));
typedef float v4f __attribute__((ext_vector_type(4)));
typedef float v8f __attribute__((ext_vector_type(8)));

#define N_NODES 100000
#define K_NBRS  32
#define D       128
// trimmed mean: keep sorted ranks [14, 18), i.e. middle 4 of 32

// ---------------------------------------------------------------------------
// Kernel 1: h = x * W^T via V_WMMA_F32_16X16X4_F32 (exact fp32 path).
// One wave computes a 16-row stripe of h (16 x 128). 8 C-tiles of 16x16,
// each accumulated with 32 WMMA ops over K=128 in steps of 4.
//
// f32 A-matrix 16x4 VGPR layout (ISA 7.12.2): lane l<16 holds rows M=l with
// K=0 (v0), K=1 (v1); lanes 16+l hold K=2 (v0), K=3 (v1). So a float2 load at
// column 4j + 2*half gives exactly one A fragment. B (4x16, K-major rows
// striped across lanes) mirrors this with W's rows as columns of B.
// ---------------------------------------------------------------------------
__global__ __launch_bounds__(256)
void gemm_xwT_wmma(const float* __restrict__ x, const float* __restrict__ W,
                   float* __restrict__ h, int n_tiles) {
  const int wave = blockIdx.x * (blockDim.x >> 5) + (threadIdx.x >> 5);
  if (wave >= n_tiles) return;
  const int lane = threadIdx.x & 31;
  const int half = lane >> 4;   // 0: lanes 0-15, 1: lanes 16-31
  const int l16  = lane & 15;
  const int n0   = wave * 16;

  // Stage A fragments (x rows for this 16-row tile) once; reuse for all 8
  // output tiles. 32 fragments x 2 VGPRs = 64 VGPRs per lane.
  const v2f* xrow = (const v2f*)(x + (size_t)(n0 + l16) * D);
  __builtin_prefetch(xrow, 0, 0);   // global_prefetch_b8
  v2f afrag[32];
#pragma unroll
  for (int j = 0; j < 32; ++j) afrag[j] = xrow[2 * j + half];

#pragma unroll
  for (int ot = 0; ot < 8; ++ot) {
    const int o0 = ot * 16;
    const v2f* wrow = (const v2f*)(W + (size_t)(o0 + l16) * D);
    v8f c = {};
#pragma unroll
    for (int j = 0; j < 32; ++j) {
      v2f b = wrow[2 * j + half];
      // 8 args: (neg_a, A, neg_b, B, c_mod, C, reuse_a, reuse_b)
      c = __builtin_amdgcn_wmma_f32_16x16x4_f32(
              false, afrag[j], false, b, (short)0, c, false, false);
    }
    // C/D layout: VGPR r, lanes 0-15 -> (M=r, N=l16); lanes 16-31 -> (M=r+8).
    float* hout = h + (size_t)(n0 + half * 8) * D + o0 + l16;
#pragma unroll
    for (int r = 0; r < 8; ++r) hout[(size_t)r * D] = c[r];
  }
}

// ---------------------------------------------------------------------------
// Kernel 2: gather 32 neighbor rows per node, per-feature trimmed mean of the
// middle 4 via a fully-unrolled 32-element bitonic network in registers.
//
// Each thread owns one node and a group of 4 consecutive features (float4
// gathers -> global_load_b128, 4x fewer load instructions for the same
// bytes). A wave = one node x 32 feature-groups: the neighbor index is
// wave-uniform, and each (n,k) gather covers one contiguous 512 B row segment
// of h (L2-resident: 51.2 MB < 192 MB L2). Block = 256 threads = 8 nodes.
// ---------------------------------------------------------------------------
__global__ __launch_bounds__(256)
void trimmed_gather(const float* __restrict__ h, const int* __restrict__ nbrs,
                    float* __restrict__ out) {
  __shared__ int sidx[8 * K_NBRS];
  const int nblock = blockIdx.x * 8;
  const int tid = threadIdx.x;
  {
    const int n = nblock + (tid >> 5);
    const int k = tid & 31;
    sidx[tid] = (n < N_NODES) ? nbrs[(size_t)n * K_NBRS + k] : 0;
  }
  __syncthreads();

  const int local_n = tid >> 5;     // 0..7: node within block (one per wave)
  const int o4      = tid & 31;     // feature group: features [4*o4, 4*o4+4)
  const int n       = nblock + local_n;
  if (n >= N_NODES) return;

  v4f v[32];                         // 128 VGPRs of gathered values
#pragma unroll
  for (int k = 0; k < 32; ++k) {
    const int idx = sidx[local_n * K_NBRS + k];   // wave-uniform
    v[k] = *(const v4f*)(h + (size_t)idx * D + o4 * 4);
  }

  // Bitonic sort (ascending), applied component-wise to all 4 features,
  // fully unrolled so everything stays in VGPRs.
#pragma unroll
  for (int kk = 2; kk <= 32; kk <<= 1) {
#pragma unroll
    for (int j = kk >> 1; j > 0; j >>= 1) {
#pragma unroll
      for (int i = 0; i < 32; ++i) {
        const int l = i ^ j;
        if (l > i) {
          const bool up = ((i & kk) == 0);
          const v4f a = v[i], b = v[l];
          v4f mn, mx;
#pragma unroll
          for (int c = 0; c < 4; ++c) {
            mn[c] = fminf(a[c], b[c]);
            mx[c] = fmaxf(a[c], b[c]);
          }
          v[i] = up ? mn : mx;
          v[l] = up ? mx : mn;
        }
      }
    }
  }

  v4f r;
#pragma unroll
  for (int c = 0; c < 4; ++c)
    r[c] = (v[14][c] + v[15][c] + v[16][c] + v[17][c]) * 0.25f;
  *(v4f*)(out + (size_t)n * D + o4 * 4) = r;
}

// ---------------------------------------------------------------------------
extern "C" void kernel_launch(void* const* d_in, const int* in_sizes, int n_in,
                              void* d_out, int out_size, void* d_ws, size_t ws_size,
                              hipStream_t stream) {
  const float* x    = (const float*)d_in[0];   // (N, 128) fp32
  const float* W    = (const float*)d_in[1];   // (128, 128) fp32
  const int*   nbrs = (const int*)d_in[2];     // (N, 32) int32
  float* out = (float*)d_out;                  // (N, 128) fp32
  float* h   = (float*)d_ws;                   // scratch: N*128*4 = 51.2 MB

  const int n_tiles = N_NODES / 16;            // 6250 (exact, no tail)
  const int waves_per_block = 256 / 32;        // 8
  const int gemm_blocks = (n_tiles + waves_per_block - 1) / waves_per_block;
  gemm_xwT_wmma<<<gemm_blocks, 256, 0, stream>>>(x, W, h, n_tiles);

  const int gather_blocks = (N_NODES + 7) / 8; // 8 nodes per block
  trimmed_gather<<<gather_blocks, 256, 0, stream>>>(h, nbrs, out);
}